// LeastSquareModule_807453852437
// MI455X (gfx1250) — compile-verified
//
#include <hip/hip_runtime.h>
#include <math.h>

// ---------------------------------------------------------------------------
// Least-squares surface normals from depth, 9x9 masked patch per pixel.
// B=4, H=W=256, f32 in / f32 out.
// VALU-bound stencil: LDS-tiled depth, async global->LDS b128/b32 (CDNA5,
// ASYNCcnt), fully unrolled FMA accumulation of the symmetric 3x3 Gram matrix.
// ---------------------------------------------------------------------------

namespace {
constexpr int HH    = 256;
constexpr int WW    = 256;
constexpr int TILE  = 16;
constexpr int HALO  = 4;                 // (9-1)/2
constexpr int PAD   = TILE + 2 * HALO;   // 24
constexpr float GAMMA_T = 0.15f;
constexpr float EPS_DET = 1e-5f;
}

__device__ __forceinline__ int clampi(int v, int lo, int hi) {
    return v < lo ? lo : (v > hi ? hi : v);
}

// CDNA5 async global->LDS copy path (ASYNCcnt), with portable fallbacks.
#if defined(__HIP_DEVICE_COMPILE__) && __has_builtin(__builtin_amdgcn_global_load_async_to_lds_b32)
#define USE_ASYNC_LDS 1
#else
#define USE_ASYNC_LDS 0
#endif
#if defined(__HIP_DEVICE_COMPILE__) && __has_builtin(__builtin_amdgcn_global_load_async_to_lds_b128)
#define USE_ASYNC_LDS_B128 1
#else
#define USE_ASYNC_LDS_B128 0
#endif

typedef int vint4 __attribute__((vector_size(16)));
typedef __attribute__((address_space(1))) int   gint_t;    // global (AS1) scalar
typedef __attribute__((address_space(3))) int   lint_t;    // LDS    (AS3) scalar
typedef __attribute__((address_space(1))) vint4 gint4_t;   // global (AS1) 4xint
typedef __attribute__((address_space(3))) vint4 lint4_t;   // LDS    (AS3) 4xint

__device__ __forceinline__ void wait_async_all() {
#if defined(__HIP_DEVICE_COMPILE__)
#if __has_builtin(__builtin_amdgcn_s_wait_asynccnt)
    __builtin_amdgcn_s_wait_asynccnt(0);
#else
    asm volatile("s_wait_asynccnt 0" ::: "memory");
#endif
#endif
}

__global__ __launch_bounds__(256)
void lstsq_normals_kernel(const float* __restrict__ depth,   // [B,1,H,W]
                          const float* __restrict__ fov,     // [B]
                          float* __restrict__ out)           // [B,3,H,W]
{
    __shared__ float tile[PAD * PAD];    // 2304 B of 320 KB LDS: depth (z) only

    const int tid = threadIdx.x;
    const int tx  = tid & (TILE - 1);
    const int ty  = tid >> 4;
    const int ox  = blockIdx.x * TILE;   // tile origin column
    const int oy  = blockIdx.y * TILE;   // tile origin row
    const int b   = blockIdx.z;

    const float* dbase = depth + (size_t)b * (HH * WW);

    // ---- stage depth tile (edge-clamped) into LDS ------------------------
    // Column clamping only happens in the first/last block column; elsewhere
    // rows are contiguous in global memory and we can stream 16B per lane.
    const bool col_interior = (ox >= HALO) && (ox + TILE + HALO <= WW);

#if USE_ASYNC_LDS && USE_ASYNC_LDS_B128
    if (col_interior) {
        // 576 floats = 144 lanes x float4; row stride 24 keeps 16B alignment.
        if (tid < (PAD * PAD) / 4) {
            const int lr  = tid / (PAD / 4);            // 0..23
            const int lc4 = (tid % (PAD / 4)) * 4;      // 0,4,...,20
            const int gr  = clampi(oy - HALO + lr, 0, HH - 1);
            const float* gp = dbase + gr * WW + (ox - HALO + lc4);
            __builtin_amdgcn_global_load_async_to_lds_b128(
                (gint4_t*)gp, (lint4_t*)&tile[lr * PAD + lc4], 0, 0);
        }
    } else
#endif
    {
#pragma unroll
        for (int it = 0; it < 3; ++it) {
            const int k = tid + it * 256;
            if (k < PAD * PAD) {
                const int lr = k / PAD;
                const int lc = k - lr * PAD;
                const int gr = clampi(oy - HALO + lr, 0, HH - 1);
                const int gc = clampi(ox - HALO + lc, 0, WW - 1);
                const float* gp = dbase + gr * WW + gc;
#if USE_ASYNC_LDS
                __builtin_amdgcn_global_load_async_to_lds_b32(
                    (gint_t*)gp, (lint_t*)&tile[k], 0, 0);
#else
                tile[k] = *gp;
#endif
            }
        }
    }
#if USE_ASYNC_LDS
    wait_async_all();      // this wave's async copies have landed in LDS
#endif
    __syncthreads();       // all waves' copies visible

    // ---- per-pixel coordinate scales ------------------------------------
    // x = z*(c-127.5)*t/W, y = z*(r-127.5)*t/H, t = 2*tan(fov/2)
    const float t_fov = 2.0f * tanf(fov[b] * 0.5f);
    const float scale = t_fov * (1.0f / 256.0f);

    const int row = oy + ty;
    const int col = ox + tx;

    float sx[9], sy[9];
#pragma unroll
    for (int j = 0; j < 9; ++j) {
        const int gc = clampi(col - HALO + j, 0, WW - 1);
        const int gr = clampi(row - HALO + j, 0, HH - 1);
        sx[j] = ((float)gc - 127.5f) * scale;
        sy[j] = ((float)gr - 127.5f) * scale;
    }

    const float zc  = tile[(ty + HALO) * PAD + (tx + HALO)];
    const float thr = GAMMA_T * zc;    // |z - zc| > gamma*zc  <=>  |dz/zc| > gamma (zc > 0)

    // ---- accumulate symmetric Gram matrix + RHS over 81 taps -------------
    float a00 = 0.f, a01 = 0.f, a02 = 0.f, a11 = 0.f, a12 = 0.f, a22 = 0.f;
    float b0  = 0.f, b1  = 0.f, b2  = 0.f;

#pragma unroll
    for (int i = 0; i < 9; ++i) {
        const float  syi = sy[i];
        const float* rp  = &tile[(ty + i) * PAD + tx];
#pragma unroll
        for (int j = 0; j < 9; ++j) {
            const float z  = rp[j];
            const float zm = (fabsf(z - zc) > thr) ? 0.0f : z;  // mask all 3 channels
            const float x  = zm * sx[j];
            const float y  = zm * syi;
            a00 = fmaf(x, x,  a00);
            a01 = fmaf(x, y,  a01);
            a02 = fmaf(x, zm, a02);
            a11 = fmaf(y, y,  a11);
            a12 = fmaf(y, zm, a12);
            a22 = fmaf(zm, zm, a22);
            b0 += x; b1 += y; b2 += zm;
        }
    }

    // ---- solve A n = b via adjugate; det < EPS -> A := I -----------------
    const float c00 = fmaf(a11, a22, -a12 * a12);
    const float c01 = fmaf(a02, a12, -a01 * a22);
    const float c02 = fmaf(a01, a12, -a02 * a11);
    const float det = a00 * c00 + a01 * c01 + a02 * c02;

    float n0, n1, n2;
    if (det < EPS_DET) {
        n0 = b0; n1 = b1; n2 = b2;           // inv(I) @ Atb
    } else {
        const float c11 = fmaf(a00, a22, -a02 * a02);
        const float c12 = fmaf(a01, a02, -a00 * a12);
        const float c22 = fmaf(a00, a11, -a01 * a01);
        const float inv = 1.0f / det;
        n0 = (c00 * b0 + c01 * b1 + c02 * b2) * inv;
        n1 = (c01 * b0 + c11 * b1 + c12 * b2) * inv;
        n2 = (c02 * b0 + c12 * b1 + c22 * b2) * inv;
    }

    // ---- normalize, NaN -> 0, negate -------------------------------------
    const float nl2 = n0 * n0 + n1 * n1 + n2 * n2;
    float o0 = 0.f, o1 = 0.f, o2 = 0.f;
    if (nl2 > 0.0f && isfinite(nl2)) {
        const float s = rsqrtf(nl2);
        o0 = -n0 * s; o1 = -n1 * s; o2 = -n2 * s;
        if (isnan(o0)) o0 = 0.f;
        if (isnan(o1)) o1 = 0.f;
        if (isnan(o2)) o2 = 0.f;
    }

    const size_t plane = (size_t)HH * WW;
    const size_t base  = (size_t)b * 3 * plane + (size_t)row * WW + col;
    out[base]             = o0;
    out[base + plane]     = o1;
    out[base + 2 * plane] = o2;
}

extern "C" void kernel_launch(void* const* d_in, const int* in_sizes, int n_in,
                              void* d_out, int out_size, void* d_ws, size_t ws_size,
                              hipStream_t stream) {
    const float* depth = (const float*)d_in[0];   // [B,1,256,256] f32
    const float* fov   = (const float*)d_in[1];   // [B] f32
    float*       out   = (float*)d_out;           // [B,3,256,256] f32

    const int B = (n_in > 1) ? in_sizes[1] : 4;   // fov element count == batch
    dim3 grid(WW / TILE, HH / TILE, B);
    lstsq_normals_kernel<<<grid, 256, 0, stream>>>(depth, fov, out);
}